// CoverageLoss_42769284334028
// MI455X (gfx1250) — compile-verified
//
#include <hip/hip_runtime.h>

typedef __attribute__((ext_vector_type(2))) float v2f;
typedef __attribute__((ext_vector_type(8))) float v8f;

#define WAVES_PER_BLOCK 8   // 256 threads = 8 wave32

// ---------------------------------------------------------------------------
// Shared epilogue: N-column min reduction, inside-any-box mask, partial sum.
// dmin layout (C/D f32 16x16): VGPR i = rows M=i (lanes 0-15) / M=i+8 (16-31).
// ---------------------------------------------------------------------------
__device__ __forceinline__ void coverage_epilogue(
    v8f dmin, float fx, float fy,
    const float* __restrict__ pred, int NB,
    float* __restrict__ partial, int wave, int lane)
{
    // inside-any-box flag for fragment (lane&15); inclusive comparisons as in
    // the reference (diff_lo >= 0 && diff_hi >= 0).
    float ins = 0.0f;
    for (int b = 0; b < NB; ++b) {
        const float lox = pred[b * 4 + 0];
        const float loy = pred[b * 4 + 1];
        const float hix = pred[b * 4 + 2];
        const float hiy = pred[b * 4 + 3];
        const bool inside = (fx >= lox) && (fy >= loy) && (hix >= fx) && (hiy >= fy);
        ins = inside ? 1.0f : ins;
    }

    // Min over the 16 columns: xor masks 1,2,4,8 never cross the 16-lane half
    // boundary, so rows M=i and M=i+8 reduce independently.
#pragma unroll
    for (int i = 0; i < 8; ++i) {
        float x = dmin[i];
        x = fminf(x, __shfl_xor(x, 1, 32));
        x = fminf(x, __shfl_xor(x, 2, 32));
        x = fminf(x, __shfl_xor(x, 4, 32));
        x = fminf(x, __shfl_xor(x, 8, 32));
        dmin[i] = x;
    }

    float m[16];
#pragma unroll
    for (int i = 0; i < 8; ++i) {
        m[i]     = __shfl(dmin[i], 0, 32);   // row M=i   (low half)
        m[i + 8] = __shfl(dmin[i], 16, 32);  // row M=i+8 (high half)
    }
    float insv[16];
#pragma unroll
    for (int f = 0; f < 16; ++f) insv[f] = __shfl(ins, f, 32);

    if (lane == 0) {
        float s = 0.0f;
#pragma unroll
        for (int f = 0; f < 16; ++f) s += (insv[f] > 0.0f) ? 0.0f : m[f];
        partial[wave] = s;
    }
}

// ---------------------------------------------------------------------------
// Pass 1: encode every box-boundary point p as the quad [-2px, -2py, 1, |p|^2]
// so that A_row(f) . B_col(p) = ||f - p||^2 with A_row(f) = [fx, fy, |f|^2, 1].
// 25600 quads * 16 B = 409 KB -> L2-resident, reused by all 128 waves.
// ---------------------------------------------------------------------------
__global__ void coverage_encode(const float* __restrict__ pred,
                                const float* __restrict__ boundary,
                                float* __restrict__ enc,   // [NB*BP] float4
                                int NB, int BP)
{
    const int t = blockIdx.x * blockDim.x + threadIdx.x;
    if (t >= NB * BP) return;
    const int b = t / BP;
    const int p = t - b * BP;
    const float lox = pred[b * 4 + 0];
    const float loy = pred[b * 4 + 1];
    const float hix = pred[b * 4 + 2];
    const float hiy = pred[b * 4 + 3];
    const float bx = boundary[p * 2 + 0];
    const float by = boundary[p * 2 + 1];
    const float px = bx * (hix - lox) + lox;
    const float py = by * (hiy - loy) + loy;
    float4 e;
    e.x = -2.0f * px;
    e.y = -2.0f * py;
    e.z = 1.0f;
    e.w = px * px + py * py;
    ((float4*)enc)[t] = e;
}

// ---------------------------------------------------------------------------
// Pass 2 (fast path): one wave owns 16 fragment points; streams all 1600
// pre-encoded 16-point B tiles. Inner loop is load_b64 -> wmma -> min3 only.
// B tile layout (4x16 f32, 2 VGPRs): lanes 0-15 rows K=0,1 = enc.xy ;
// lanes 16-31 rows K=2,3 = enc.zw -> constant per-lane byte offset.
// ---------------------------------------------------------------------------
__global__ void __launch_bounds__(256)
coverage_main_enc(const float* __restrict__ pred,
                  const float* __restrict__ frags,
                  const float* __restrict__ enc,
                  float* __restrict__ partial,
                  int NB, int BP)
{
    const int lane  = threadIdx.x & 31;
    const int wave  = blockIdx.x * WAVES_PER_BLOCK + (threadIdx.x >> 5);
    const int fbase = wave * 16;
    const int n     = lane & 15;
    const bool low  = lane < 16;

    const float fx = frags[(fbase + n) * 2 + 0];
    const float fy = frags[(fbase + n) * 2 + 1];

    // A tile (16x4 f32): lanes 0-15 -> K=0,1 = {fx,fy}; 16-31 -> K=2,3 = {|f|^2,1}
    v2f Am;
    Am.x = low ? fx : (fx * fx + fy * fy);
    Am.y = low ? fy : 1.0f;

    v8f dmin;
#pragma unroll
    for (int i = 0; i < 8; ++i) dmin[i] = 3.0e38f;
    v8f c0 = {};

    // Per-lane element offset into the 16-float4 tile record.
    const float* __restrict__ basep = enc + (n * 4 + (low ? 0 : 2));
    const int ntiles = (NB * BP) >> 4;   // 1600
    for (int j = 0; j < ntiles; ++j) {
        v2f Bm = *(const v2f*)(basep + (size_t)j * 64);
        v8f d = __builtin_amdgcn_wmma_f32_16x16x4_f32(
            false, Am, false, Bm, (short)0, c0, false, false);
#pragma unroll
        for (int i = 0; i < 8; ++i) dmin[i] = fminf(dmin[i], d[i]);
    }

    coverage_epilogue(dmin, fx, fy, pred, NB, partial, wave, lane);
}

// ---------------------------------------------------------------------------
// Fallback (workspace too small): build B tiles in-loop as before.
// ---------------------------------------------------------------------------
__global__ void __launch_bounds__(256)
coverage_main_fused(const float* __restrict__ pred,
                    const float* __restrict__ frags,
                    const float* __restrict__ boundary,
                    float* __restrict__ partial,
                    int NB, int BP)
{
    const int lane  = threadIdx.x & 31;
    const int wave  = blockIdx.x * WAVES_PER_BLOCK + (threadIdx.x >> 5);
    const int fbase = wave * 16;
    const int n     = lane & 15;
    const bool low  = lane < 16;

    const float fx = frags[(fbase + n) * 2 + 0];
    const float fy = frags[(fbase + n) * 2 + 1];

    v2f Am;
    Am.x = low ? fx : (fx * fx + fy * fy);
    Am.y = low ? fy : 1.0f;

    v8f dmin;
#pragma unroll
    for (int i = 0; i < 8; ++i) dmin[i] = 3.0e38f;
    v8f c0 = {};

    const int nchunk = BP >> 4;
    for (int chunk = 0; chunk < nchunk; ++chunk) {
        const float bx = boundary[(chunk * 16 + n) * 2 + 0];
        const float by = boundary[(chunk * 16 + n) * 2 + 1];
        for (int b = 0; b < NB; ++b) {
            const float lox = pred[b * 4 + 0];
            const float loy = pred[b * 4 + 1];
            const float hix = pred[b * 4 + 2];
            const float hiy = pred[b * 4 + 3];
            const float px = bx * (hix - lox) + lox;
            const float py = by * (hiy - loy) + loy;
            v2f Bm;
            Bm.x = low ? (-2.0f * px) : 1.0f;
            Bm.y = low ? (-2.0f * py) : (px * px + py * py);
            v8f d = __builtin_amdgcn_wmma_f32_16x16x4_f32(
                false, Am, false, Bm, (short)0, c0, false, false);
#pragma unroll
            for (int i = 0; i < 8; ++i) dmin[i] = fminf(dmin[i], d[i]);
        }
    }

    coverage_epilogue(dmin, fx, fy, pred, NB, partial, wave, lane);
}

// Fixed-order serial sum of the per-wave partials -> deterministic scalar loss.
__global__ void coverage_reduce(const float* __restrict__ partial,
                                float* __restrict__ out,
                                int nPartial, float invFP)
{
    if (threadIdx.x == 0 && blockIdx.x == 0) {
        float s = 0.0f;
        for (int i = 0; i < nPartial; ++i) s += partial[i];
        out[0] = s * invFP;
    }
}

extern "C" void kernel_launch(void* const* d_in, const int* in_sizes, int n_in,
                              void* d_out, int out_size, void* d_ws, size_t ws_size,
                              hipStream_t stream) {
    const float* pred     = (const float*)d_in[0];  // [64,4]
    const float* frags    = (const float*)d_in[1];  // [32,64,2]
    const float* boundary = (const float*)d_in[2];  // [1,400,2]

    const int NB = in_sizes[0] / 4;   // 64 boxes
    const int NF = in_sizes[1] / 2;   // 2048 fragment points
    const int BP = in_sizes[2] / 2;   // 400 boundary samples

    const int nWaves  = NF / 16;                    // 128
    const int nBlocks = nWaves / WAVES_PER_BLOCK;   // 16

    const size_t encFloats = (size_t)NB * BP * 4;   // 102400 floats = 409.6 KB
    const size_t needed    = (encFloats + (size_t)nWaves) * sizeof(float);

    if (ws_size >= needed) {
        float* enc     = (float*)d_ws;              // [NB*BP] float4 quads
        float* partial = enc + encFloats;           // [nWaves]
        const int nEnc = NB * BP;
        coverage_encode<<<(nEnc + 255) / 256, 256, 0, stream>>>(pred, boundary, enc, NB, BP);
        coverage_main_enc<<<nBlocks, 256, 0, stream>>>(pred, frags, enc, partial, NB, BP);
        const float invFP = 1.0f / 64.0f;           // fragments.shape[1] == 64
        coverage_reduce<<<1, 32, 0, stream>>>(partial, (float*)d_out, nWaves, invFP);
    } else {
        float* partial = (float*)d_ws;              // [nWaves]
        coverage_main_fused<<<nBlocks, 256, 0, stream>>>(pred, frags, boundary, partial, NB, BP);
        const float invFP = 1.0f / 64.0f;
        coverage_reduce<<<1, 32, 0, stream>>>(partial, (float*)d_out, nWaves, invFP);
    }
}